// SeesawLoss_11330123727119
// MI455X (gfx1250) — compile-verified
//
#include <hip/hip_runtime.h>

// Seesaw loss, fp32, gfx1250 (MI455X).
// HBM-bound: ~150 MB traffic -> ~6.5us floor at 23.3 TB/s.
// wave32 reduction done with one V_WMMA_F32_16X16X4_F32 per wave.

#define NCLS      16
#define P_EXP     0.8f
#define Q_EXP     2.0f
#define LOG_EPS  -4.6051701859880913680f   // log(0.01)

#define MAIN_BLOCKS   1024
#define MAIN_THREADS  256
#define HIST_BLOCKS   1024
#define HIST_THREADS  256

typedef float v2f __attribute__((ext_vector_type(2)));
typedef float v8f __attribute__((ext_vector_type(8)));

// ---------------------------------------------------------------- init ----
__global__ void init_ws_kernel(unsigned int* __restrict__ hist) {
  if (threadIdx.x < NCLS) hist[threadIdx.x] = 0u;
}

// ----------------------------------------------------------- histogram ----
// Integer atomics are associative -> deterministic regardless of order.
__global__ void hist_kernel(const int* __restrict__ labels,
                            unsigned int* __restrict__ hist, int n) {
  __shared__ unsigned int h[NCLS];
  if (threadIdx.x < NCLS) h[threadIdx.x] = 0u;
  __syncthreads();
  const int stride = blockDim.x * gridDim.x;
  for (int i = blockIdx.x * blockDim.x + threadIdx.x; i < n; i += stride) {
    atomicAdd(&h[labels[i] & (NCLS - 1)], 1u);
  }
  __syncthreads();
  if (threadIdx.x < NCLS) atomicAdd(&hist[threadIdx.x], h[threadIdx.x]);
}

// ---------------------------------------------------------- main pass ----
__global__ void __launch_bounds__(MAIN_THREADS)
seesaw_main_kernel(const float* __restrict__ logits,
                   const int* __restrict__ labels,
                   const float* __restrict__ cum_samples,
                   const unsigned int* __restrict__ hist,
                   float* __restrict__ block_parts,   // [gridDim.x][2]
                   int n_pix) {
  constexpr int HW = 512 * 512;                 // class-plane stride (elements)
  __shared__ float s_plogcum[NCLS];
  __shared__ float s_wnll[MAIN_THREADS / 32];
  __shared__ float s_wmask[MAIN_THREADS / 32];

  const int t = threadIdx.x;
  if (t < NCLS) {
    const float cum = cum_samples[t] + (float)hist[t];
    s_plogcum[t] = P_EXP * __logf(fmaxf(cum, 1.0f));
  }
  __syncthreads();

  float plc[NCLS];
#pragma unroll
  for (int c = 0; c < NCLS; ++c) plc[c] = s_plogcum[c];

  float nll_acc = 0.0f, mask_acc = 0.0f;
  const int stride = blockDim.x * gridDim.x;
  for (int i = blockIdx.x * blockDim.x + t; i < n_pix; i += stride) {
    const int b  = i / HW;                      // HW is a power of two -> shifts
    const int hw = i - b * HW;
    const float* xp = logits + (size_t)b * NCLS * HW + hw;
    const int L = labels[i];
    if (i + stride < n_pix)
      __builtin_prefetch(&labels[i + stride], 0, 0);   // global_prefetch_b8

    // load the 16 class logits (each load coalesced across the wave),
    // track x[L] and row max with select chains (no dynamic reg indexing)
    float x[NCLS];
    float xL = 0.0f;
    float m = -3.402823466e+38f;
#pragma unroll
    for (int c = 0; c < NCLS; ++c) {
      const float v = xp[(size_t)c * HW];
      x[c] = v;
      xL = (c == L) ? v : xL;
      m = fmaxf(m, v);
    }
    float se = 0.0f;
#pragma unroll
    for (int c = 0; c < NCLS; ++c) se += __expf(x[c] - m);
    const float lse = m + __logf(se);

    const float thr = fmaxf(xL - lse, LOG_EPS);  // log(max(softmax_L, EPS))
    const float plL = s_plogcum[L];

    // adjusted logits; both seesaw terms vanish at c == L automatically
    float adj[NCLS];
    float m2 = -3.402823466e+38f;
#pragma unroll
    for (int c = 0; c < NCLS; ++c) {
      const float a = x[c] + fminf(0.0f, plc[c] - plL)
                           + Q_EXP * fmaxf(0.0f, (x[c] - lse) - thr);
      adj[c] = a;
      m2 = fmaxf(m2, a);
    }
    float se2 = 0.0f;
#pragma unroll
    for (int c = 0; c < NCLS; ++c) se2 += __expf(adj[c] - m2);
    const float nll = (m2 + __logf(se2)) - xL;   // adj[L] == x[L]

    if (L != 0) { nll_acc += nll; mask_acc += 1.0f; }
  }

  // ---- wave32 reduction via one V_WMMA_F32_16X16X4_F32 ----
  // A (16x4): VGPR0 = nll per lane (K0 lanes0-15 / K2 lanes16-31),
  //           VGPR1 = mask per lane (K1 / K3).
  // B (4x16): column selector -> cols 0-7 sum the nll rows of A (K0,K2),
  //           cols 8-15 sum the mask rows (K1,K3).
  // D[m][n<8]  = nll(m)+nll(m+16);  D[m][n>=8] = mask(m)+mask(m+16).
  const int lane = t & 31;
  v2f amat; amat.x = nll_acc; amat.y = mask_acc;
  v2f bmat;
  bmat.x = ((lane & 15) < 8) ? 1.0f : 0.0f;
  bmat.y = ((lane & 15) < 8) ? 0.0f : 1.0f;
  v8f dmat = {};
  dmat = __builtin_amdgcn_wmma_f32_16x16x4_f32(false, amat, false, bmat,
                                               (short)0, dmat, false, false);
  float tsum = dmat[0] + dmat[1] + dmat[2] + dmat[3] +
               dmat[4] + dmat[5] + dmat[6] + dmat[7];
  // fold the two 16-lane halves of the D column space
  const float u = tsum + __shfl_xor(tsum, 16, 32);
  // lane 0 now holds total nll of the wave, lane 8 holds total mask
  const float wave_nll  = __shfl(u, 0, 32);
  const float wave_mask = __shfl(u, 8, 32);

  const int wid = t >> 5;
  if (lane == 0) { s_wnll[wid] = wave_nll; s_wmask[wid] = wave_mask; }
  __syncthreads();
  if (t == 0) {
    float bn = 0.0f, bm = 0.0f;
#pragma unroll
    for (int w = 0; w < MAIN_THREADS / 32; ++w) { bn += s_wnll[w]; bm += s_wmask[w]; }
    block_parts[2 * blockIdx.x + 0] = bn;      // fixed-order, deterministic
    block_parts[2 * blockIdx.x + 1] = bm;
  }
}

// ------------------------------------------------------------ finalize ----
// Fixed-pairing tree over block partials -> bit-deterministic scalar.
__global__ void finalize_kernel(const float* __restrict__ parts, int nblocks,
                                float* __restrict__ out) {
  __shared__ float sn[256];
  __shared__ float sm[256];
  const int t = threadIdx.x;
  float an = 0.0f, am = 0.0f;
  for (int i = t; i < nblocks; i += 256) {
    an += parts[2 * i + 0];
    am += parts[2 * i + 1];
  }
  sn[t] = an; sm[t] = am;
  __syncthreads();
  for (int off = 128; off > 0; off >>= 1) {
    if (t < off) { sn[t] += sn[t + off]; sm[t] += sm[t + off]; }
    __syncthreads();
  }
  if (t == 0) out[0] = sn[0] / sm[0];
}

// -------------------------------------------------------------- launch ----
extern "C" void kernel_launch(void* const* d_in, const int* in_sizes, int n_in,
                              void* d_out, int out_size, void* d_ws, size_t ws_size,
                              hipStream_t stream) {
  const float* cls_score   = (const float*)d_in[0];
  const int*   labels      = (const int*)d_in[1];
  const float* cum_samples = (const float*)d_in[2];
  const int n_pix = in_sizes[1];               // 8*512*512 = 2,097,152

  unsigned int* hist = (unsigned int*)d_ws;                    // 16 u32
  float* parts = (float*)((char*)d_ws + 64);                   // MAIN_BLOCKS*2 f32
  float* out = (float*)d_out;

  init_ws_kernel<<<1, 32, 0, stream>>>(hist);
  hist_kernel<<<HIST_BLOCKS, HIST_THREADS, 0, stream>>>(labels, hist, n_pix);
  seesaw_main_kernel<<<MAIN_BLOCKS, MAIN_THREADS, 0, stream>>>(
      cls_score, labels, cum_samples, hist, parts, n_pix);
  finalize_kernel<<<1, 256, 0, stream>>>(parts, MAIN_BLOCKS, out);
}